// GCN_48473000902749
// MI455X (gfx1250) — compile-verified
//
#include <hip/hip_runtime.h>
#include <hip/hip_bf16.h>

typedef __attribute__((ext_vector_type(2))) float v2f;
typedef __attribute__((ext_vector_type(8))) float v8f;

#define IN_FEAT 512
#define HIDDEN  16
#define NCLS    7

// ---------------- degree / normalization ----------------

__global__ void gcn_deg_init(float* deg, int n) {
    int i = blockIdx.x * blockDim.x + threadIdx.x;
    if (i < n) deg[i] = 1.0f;   // self-loop
}

__global__ void gcn_deg_accum(const int* __restrict__ dst, float* deg, int E) {
    int e = blockIdx.x * blockDim.x + threadIdx.x;
    if (e < E) atomicAdd(&deg[dst[e]], 1.0f);
}

__global__ void gcn_deg_rsqrt(float* dinv, int n) {
    int i = blockIdx.x * blockDim.x + threadIdx.x;
    if (i < n) {
        float d = dinv[i];
        dinv[i] = (d > 0.0f) ? rsqrtf(d) : 0.0f;
    }
}

__global__ void gcn_zero(float* p, int n) {
    int i = blockIdx.x * blockDim.x + threadIdx.x;
    if (i < n) p[i] = 0.0f;
}

// ---------------- GEMM1: h = x @ W1  (f32 WMMA, 16x16x4) ----------------
// One wave per 16-row tile. A: x[tile rows, K], B: W1[K, 16], C: 16x16 f32.

__global__ void gcn_gemm1_wmma(const float* __restrict__ x,
                               const float* __restrict__ W1,
                               float* __restrict__ h, int n) {
    int wave = (blockIdx.x * blockDim.x + threadIdx.x) >> 5;
    int lane = threadIdx.x & 31;
    int row0 = wave * 16;
    if (row0 >= n) return;                 // wave-uniform exit

    int r    = lane & 15;                  // column of C / row-in-tile of A
    int hi   = lane >> 4;                  // 0: K+{0,1}, 1: K+{2,3}
    int koff = hi * 2;
    int row  = row0 + r;
    if (row >= n) row = n - 1;             // clamp (lane-safe, no EXEC change at wmma)
    const float* xrow = x + (size_t)row * IN_FEAT;

    v8f c = {0.f, 0.f, 0.f, 0.f, 0.f, 0.f, 0.f, 0.f};

    #pragma unroll 4
    for (int k = 0; k < IN_FEAT; k += 4) {
        if ((k & 63) == 0)                 // wave-uniform: prefetch ahead in row
            __builtin_prefetch(xrow + k + 128, 0, 1);
        v2f a = *(const v2f*)(xrow + k + koff);   // 8B-aligned (k,koff even)
        int kk = k + koff;
        v2f b;
        b.x = W1[(size_t)kk * HIDDEN + r];
        b.y = W1[(size_t)(kk + 1) * HIDDEN + r];
        c = __builtin_amdgcn_wmma_f32_16x16x4_f32(false, a, false, b,
                                                  (short)0, c, false, false);
    }

    int srow = row0 + hi * 8;
    #pragma unroll
    for (int i = 0; i < 8; ++i) {
        int rr = srow + i;
        if (rr < n) h[(size_t)rr * HIDDEN + r] = c[i];
    }
}

// ---------------- edge aggregation, layer 1 (16 features) ----------------
// thread t: edge e = t/4, feature quad q = t%4

__global__ void gcn_agg1(const int* __restrict__ src, const int* __restrict__ dst,
                         const float* __restrict__ dinv, const float* __restrict__ h,
                         float* out1, int E) {
    int t = blockIdx.x * blockDim.x + threadIdx.x;
    int e = t >> 2, q = t & 3;
    if (e >= E) return;
    int s = src[e], d = dst[e];
    float w = dinv[s] * dinv[d];
    float4 v = *((const float4*)(h + (size_t)s * HIDDEN) + q);
    float* o = out1 + (size_t)d * HIDDEN + q * 4;
    atomicAdd(o + 0, v.x * w);
    atomicAdd(o + 1, v.y * w);
    atomicAdd(o + 2, v.z * w);
    atomicAdd(o + 3, v.w * w);
}

// ---------------- self-loop + bias + relu ----------------

__global__ void gcn_selfloop_relu(const float* __restrict__ out1,
                                  const float* __restrict__ h,
                                  const float* __restrict__ dinv,
                                  const float* __restrict__ b1,
                                  float* hrelu, int n) {
    int i = blockIdx.x * blockDim.x + threadIdx.x;
    if (i >= n) return;
    float w = dinv[i] * dinv[i];
    const float* o = out1 + (size_t)i * HIDDEN;
    const float* hh = h + (size_t)i * HIDDEN;
    float* dst = hrelu + (size_t)i * HIDDEN;
    #pragma unroll
    for (int c = 0; c < HIDDEN; ++c)
        dst[c] = fmaxf(o[c] + hh[c] * w + b1[c], 0.0f);
}

// ---------------- GEMM2: g = hrelu @ W2  (W2 zero-padded 16x7 -> 16x16) ----------------

__global__ void gcn_gemm2_wmma(const float* __restrict__ hrelu,
                               const float* __restrict__ W2,
                               float* __restrict__ g, int n) {
    int wave = (blockIdx.x * blockDim.x + threadIdx.x) >> 5;
    int lane = threadIdx.x & 31;
    int row0 = wave * 16;
    if (row0 >= n) return;

    int r    = lane & 15;
    int hi   = lane >> 4;
    int koff = hi * 2;
    int row  = row0 + r;
    if (row >= n) row = n - 1;
    const float* hrow = hrelu + (size_t)row * HIDDEN;

    v8f c = {0.f, 0.f, 0.f, 0.f, 0.f, 0.f, 0.f, 0.f};

    #pragma unroll
    for (int k = 0; k < HIDDEN; k += 4) {
        int kk = k + koff;
        v2f a = *(const v2f*)(hrow + kk);
        v2f b;
        b.x = (r < NCLS) ? W2[(size_t)kk * NCLS + r] : 0.0f;
        b.y = (r < NCLS) ? W2[(size_t)(kk + 1) * NCLS + r] : 0.0f;
        c = __builtin_amdgcn_wmma_f32_16x16x4_f32(false, a, false, b,
                                                  (short)0, c, false, false);
    }

    int srow = row0 + hi * 8;
    #pragma unroll
    for (int i = 0; i < 8; ++i) {
        int rr = srow + i;
        if (rr < n) g[(size_t)rr * HIDDEN + r] = c[i];
    }
}

// ---------------- edge aggregation, layer 2 (7 features, out2 stride 8) ----------------

__global__ void gcn_agg2(const int* __restrict__ src, const int* __restrict__ dst,
                         const float* __restrict__ dinv, const float* __restrict__ g,
                         float* out2, int E) {
    int e = blockIdx.x * blockDim.x + threadIdx.x;
    if (e >= E) return;
    int s = src[e], d = dst[e];
    float w = dinv[s] * dinv[d];
    const float* gs = g + (size_t)s * HIDDEN;
    float* o = out2 + (size_t)d * 8;
    #pragma unroll
    for (int c = 0; c < NCLS; ++c)
        atomicAdd(o + c, gs[c] * w);
}

// ---------------- self-loop + bias + log_softmax ----------------

__global__ void gcn_logsoftmax(const float* __restrict__ out2,
                               const float* __restrict__ g,
                               const float* __restrict__ dinv,
                               const float* __restrict__ b2,
                               float* __restrict__ out, int n) {
    int i = blockIdx.x * blockDim.x + threadIdx.x;
    if (i >= n) return;
    float w = dinv[i] * dinv[i];
    const float* o = out2 + (size_t)i * 8;
    const float* gg = g + (size_t)i * HIDDEN;
    float v[NCLS];
    float m = -3.4e38f;
    #pragma unroll
    for (int c = 0; c < NCLS; ++c) {
        v[c] = o[c] + gg[c] * w + b2[c];
        m = fmaxf(m, v[c]);
    }
    float s = 0.0f;
    #pragma unroll
    for (int c = 0; c < NCLS; ++c) s += expf(v[c] - m);
    float l = logf(s);
    float* dsto = out + (size_t)i * NCLS;
    #pragma unroll
    for (int c = 0; c < NCLS; ++c) dsto[c] = v[c] - m - l;
}

// ---------------- launcher ----------------

extern "C" void kernel_launch(void* const* d_in, const int* in_sizes, int n_in,
                              void* d_out, int out_size, void* d_ws, size_t ws_size,
                              hipStream_t stream) {
    const float* x  = (const float*)d_in[0];
    const int*   ei = (const int*)d_in[1];
    const float* W1 = (const float*)d_in[2];
    const float* b1 = (const float*)d_in[3];
    const float* W2 = (const float*)d_in[4];
    const float* b2 = (const float*)d_in[5];
    float* out = (float*)d_out;

    const int n = in_sizes[0] / IN_FEAT;
    const int E = in_sizes[1] / 2;
    const int* src = ei;
    const int* dst = ei + E;

    // workspace layout (floats): dinv[n] | h[16n] (reused: out2) | out1[16n] (reused: g) | hrelu[16n]
    float* f     = (float*)d_ws;
    float* dinv  = f;
    float* h     = f + (size_t)n;
    float* out1  = f + (size_t)17 * n;
    float* hrelu = f + (size_t)33 * n;
    float* g     = out1;   // alias: out1 dead after selfloop_relu
    float* out2  = h;      // alias: h dead after selfloop_relu

    const int B = 256;
    const int tiles = (n + 15) / 16;
    const int gemmBlocks = (tiles + 7) / 8;   // 8 waves (tiles) per 256-thread block

    gcn_deg_init   <<<(n + B - 1) / B, B, 0, stream>>>(dinv, n);
    gcn_deg_accum  <<<(E + B - 1) / B, B, 0, stream>>>(dst, dinv, E);
    gcn_deg_rsqrt  <<<(n + B - 1) / B, B, 0, stream>>>(dinv, n);

    gcn_zero       <<<((size_t)16 * n + B - 1) / B, B, 0, stream>>>(out1, 16 * n);
    gcn_gemm1_wmma <<<gemmBlocks, B, 0, stream>>>(x, W1, h, n);
    gcn_agg1       <<<((size_t)4 * E + B - 1) / B, B, 0, stream>>>(src, dst, dinv, h, out1, E);
    gcn_selfloop_relu<<<(n + B - 1) / B, B, 0, stream>>>(out1, h, dinv, b1, hrelu, n);

    gcn_zero       <<<((size_t)8 * n + B - 1) / B, B, 0, stream>>>(out2, 8 * n);
    gcn_gemm2_wmma <<<gemmBlocks, B, 0, stream>>>(hrelu, W2, g, n);
    gcn_agg2       <<<(E + B - 1) / B, B, 0, stream>>>(src, dst, dinv, g, out2, E);
    gcn_logsoftmax <<<(n + B - 1) / B, B, 0, stream>>>(out2, g, dinv, b2, out, n);
}